// ZeroUpsampling_38182259261482
// MI455X (gfx1250) — compile-verified
//
#include <hip/hip_runtime.h>

// Zero-insertion 2x upsampling:
//   in : (16, 64, 128, 128) fp32
//   out: (16, 64, 256, 256) fp32, out[:, :, ::2, ::2] = x, rest = 0
//
// Pure streaming op: 64 MB read + 256 MB write, 0 FLOPs -> ~13.7 us floor
// at 23.3 TB/s. Kernel design minimizes dynamic instruction overhead per
// byte: each thread owns one float4-lane column across 4 input rows and
// writes 8 output rows' chunks (4 interleaved-value rows + 4 zero rows),
// 128 B stored per thread, all with immediate-offset NT b128 stores.
// No branches, perfect per-instruction wave32 coalescing (16 B lane stride).

typedef __attribute__((ext_vector_type(4))) float v4f;
typedef __attribute__((ext_vector_type(2))) float v2f;

// Geometry (fixed by the reference)
#define IN_H       128
#define IN_W       128
#define OUT_W      256                  // IN_W * 2
#define IN_PLANE   (IN_H * IN_W)        // 16384
#define OUT_PLANE  (OUT_W * OUT_W)      // 65536
#define OW4        64                   // OUT_W / 4 float4 chunks per out row
#define RPT        4                    // input rows per thread

__global__ __launch_bounds__(256)
void ZeroUpsampling_38182259261482_kernel(const float* __restrict__ x,
                                          float* __restrict__ out,
                                          int n_threads) {
    int tid = blockIdx.x * blockDim.x + threadIdx.x;
    if (tid >= n_threads) return;

    // tid -> (plane, row-group, ow4). Low 6 bits = ow4 keeps per-instruction
    // stores dense across the wave (lane stride 16 B).
    int ow4   = tid & (OW4 - 1);
    int hblk  = (tid >> 6) & 31;        // IN_H / RPT = 32 row groups
    int plane = tid >> 11;              // 64 * 32 = 2048 threads per plane
    int h0    = hblk * RPT;

    const float* src = x   + (size_t)plane * IN_PLANE
                           + (size_t)h0 * IN_W + ow4 * 2;
    float*       dst = out + (size_t)plane * OUT_PLANE
                           + (size_t)(2 * h0) * OUT_W + ow4 * 4;

    const v4f z = (v4f)(0.0f);

#pragma unroll
    for (int k = 0; k < RPT; ++k) {
        // Input chunk: x[plane, h0+k, 2*ow4 .. 2*ow4+1]
        v2f a = __builtin_nontemporal_load((const v2f*)(src + k * IN_W));
        v4f v = (v4f){a.x, 0.0f, a.y, 0.0f};
        // Even output row 2*(h0+k): interleaved values; odd row: zeros.
        __builtin_nontemporal_store(v, (v4f*)(dst + (2 * k) * OUT_W));
        __builtin_nontemporal_store(z, (v4f*)(dst + (2 * k + 1) * OUT_W));
    }
}

extern "C" void kernel_launch(void* const* d_in, const int* in_sizes, int n_in,
                              void* d_out, int out_size, void* d_ws, size_t ws_size,
                              hipStream_t stream) {
    const float* x = (const float*)d_in[0];
    float* out = (float*)d_out;

    // Each thread writes 8 float4s = 32 output floats.
    int n_threads = out_size / 32;          // 2,097,152
    const int block = 256;                  // 8 waves (wave32) per block
    int grid = (n_threads + block - 1) / block;

    ZeroUpsampling_38182259261482_kernel<<<grid, block, 0, stream>>>(x, out, n_threads);
}